// exVDPMLP_42855183679903
// MI455X (gfx1250) — compile-verified
//
#include <hip/hip_runtime.h>
#include <math.h>

// Problem dims (fixed by the reference)
#define BATCH 256
#define DIN   784
#define DH    512
#define DOUT  10

// LDS x-tile row stride: 788 words -> 16B-aligned row starts (788*4 = 197*16)
// and conflict-free 16-row column reads (788 mod 64 = 20, gcd(20,64)=4 -> 16
// distinct banks across the 16 rows).
#define XPAD  788

typedef __attribute__((ext_vector_type(2))) float v2f;
typedef __attribute__((ext_vector_type(8))) float v8f;
// exact pointee type expected by __builtin_amdgcn_global_load_async_to_lds_b128
typedef int v4i_vs __attribute__((vector_size(4 * sizeof(int))));

__device__ __forceinline__ float softplus_f(float x) {
    return logf(1.0f + expf(x));   // matches jnp.log(1+exp(x)) exactly
}

// ---------------------------------------------------------------------------
// Kernel 1: materialize Ws1=softplus(w_sigma1) (512x784), Ws2=softplus(w_sigma2)
// (10x512) and reduce both KL terms in one pass.
// kl = 0.5*[ sum(sp(s1) - log sp(s1)) + sum(mu1^2) - DH*DIN
//          + sum(sp(s2) - log sp(s2)) + sum(mu2^2) - DOUT*DH ]
// ---------------------------------------------------------------------------
__global__ __launch_bounds__(256) void prep_kl_kernel(
    const float* __restrict__ w_mu1, const float* __restrict__ w_sigma1,
    const float* __restrict__ w_mu2, const float* __restrict__ w_sigma2,
    float* __restrict__ ws1, float* __restrict__ ws2,
    float* __restrict__ kl_out)
{
    const int i = blockIdx.x * blockDim.x + threadIdx.x;
    float local = 0.0f;

    if (i < DH * DIN) {
        float s = softplus_f(w_sigma1[i]);
        ws1[i] = s;
        float m = w_mu1[i];
        local += (s - logf(s)) + m * m;
    }
    if (i < DOUT * DH) {
        float s = softplus_f(w_sigma2[i]);
        ws2[i] = s;
        float m = w_mu2[i];
        local += (s - logf(s)) + m * m;
    }
    if (i == 0) {
        local -= (float)(DH * DIN + DOUT * DH);   // the "- size_in" terms
    }
    local *= 0.5f;

    // wave32 xor reduction, one atomic per wave
    #pragma unroll
    for (int off = 16; off > 0; off >>= 1)
        local += __shfl_xor(local, off, 32);
    if ((threadIdx.x & 31) == 0)
        atomicAdd(kl_out, local);
}

// ---------------------------------------------------------------------------
// Kernel 2: layer-1 dual GEMM via V_WMMA_F32_16X16X4_F32.
//   mu1[b,h]   = sum_k x[b,k]   * w_mu1[k,h]  + b_mu1[h]
//   quad1[b,h] = sum_k x[b,k]^2 * Ws1[h,k]
// fused ReLU epilogue:
//   m2[b,h] = max(mu1,0)
//   d2[b,h] = (mu1>0) ? quad1 + softplus(b_sigma1[h]) : 0   (diag of sigma2)
//
// One wave per 16(batch)x16(hidden) tile, 4 waves/block sharing one batch
// tile. The shared 16x784 x-tile is staged ONCE per block into LDS with
// GLOBAL_LOAD_ASYNC_TO_LDS_B128 (ASYNCcnt) when available, then A-fragments
// are read from LDS (ds path) while B-fragments stream from global.
// ---------------------------------------------------------------------------
__global__ __launch_bounds__(128) void layer1_wmma_kernel(
    const float* __restrict__ x,       // (256,784)
    const float* __restrict__ w_mu1,   // (784,512)
    const float* __restrict__ ws1,     // (512,784)  softplus'd
    const float* __restrict__ b_mu1,   // (512)
    const float* __restrict__ b_sigma1,// (512)
    float* __restrict__ m2,            // (256,512)
    float* __restrict__ d2)            // (256,512)
{
    const int lane  = threadIdx.x;                   // 0..31
    const int htile = blockIdx.x * 4 + threadIdx.y;  // 0..31
    const int btile = blockIdx.y;                    // 0..15
    const int ll = lane & 15;
    const int lh = lane >> 4;               // 0: K={0,1}, 1: K={2,3}

    const int col = htile * 16 + ll;        // hidden col (B-matrix N)

    __shared__ float sX[16][XPAD];          // 50.4 KB of the 320 KB WGP LDS
    const int tid = threadIdx.y * 32 + threadIdx.x; // 0..127

#if __has_builtin(__builtin_amdgcn_global_load_async_to_lds_b128)
    // Async DMA global -> LDS, 16 bytes per lane-op, no VGPR round trip.
    for (int idx = tid; idx < 16 * (DIN / 4); idx += 128) {
        const int r  = idx / (DIN / 4);
        const int c4 = idx % (DIN / 4);
        const float* g = x + (size_t)(btile * 16 + r) * DIN + c4 * 4;
        float* l = &sX[r][c4 * 4];
        __builtin_amdgcn_global_load_async_to_lds_b128(
            (__attribute__((address_space(1))) v4i_vs*)(uintptr_t)g,
            (__attribute__((address_space(3))) v4i_vs*)(uintptr_t)l,
            0, 0);
    }
    asm volatile("s_wait_asynccnt 0x0" ::: "memory");
    __syncthreads();
#else
    for (int idx = tid; idx < 16 * DIN; idx += 128) {
        const int r = idx / DIN;
        const int c = idx % DIN;
        sX[r][c] = x[(size_t)(btile * 16 + r) * DIN + c];
    }
    __syncthreads();
#endif

    v8f acc_mu = {};
    v8f acc_q  = {};

    const float* wsr = ws1 + (size_t)col * DIN;

    for (int k0 = 0; k0 < DIN; k0 += 4) {
        const int k = k0 + 2 * lh;

        if (((k0 & 63) == 0) && (k0 + 64 < DIN)) {   // in-range only
            __builtin_prefetch(w_mu1 + (size_t)(k0 + 64) * DH + col, 0, 0);
            __builtin_prefetch(wsr + k0 + 64, 0, 0);
        }

        // A fragment (16x4 f32) from LDS: lane ll holds row M=ll, K=2*lh+{0,1}
        v2f a_mu;
        a_mu.x = sX[ll][k];
        a_mu.y = sX[ll][k + 1];
        v2f a_sq = a_mu * a_mu;

        // B fragment (4x16 f32): lane ll holds col N=ll, K = 2*lh + {0,1}
        v2f b_mu;
        b_mu.x = w_mu1[(size_t)k * DH + col];
        b_mu.y = w_mu1[(size_t)(k + 1) * DH + col];
        v2f b_ws;                       // B2[k,n] = Ws1[n,k]
        b_ws.x = wsr[k];
        b_ws.y = wsr[k + 1];

        acc_mu = __builtin_amdgcn_wmma_f32_16x16x4_f32(
            false, a_mu, false, b_mu, (short)0, acc_mu, false, false);
        acc_q  = __builtin_amdgcn_wmma_f32_16x16x4_f32(
            false, a_sq, false, b_ws, (short)0, acc_q, false, false);
    }

    // Epilogue. C/D layout: VGPR r -> M = r + 8*lh, N = ll.
    const float bm = b_mu1[col];
    const float bs = softplus_f(b_sigma1[col]);
    #pragma unroll
    for (int r = 0; r < 8; ++r) {
        const int b = btile * 16 + r + 8 * lh;
        const float mu = acc_mu[r] + bm;
        const size_t idx = (size_t)b * DH + col;
        m2[idx] = fmaxf(mu, 0.0f);
        d2[idx] = (mu > 0.0f) ? (acc_q[r] + bs) : 0.0f;
    }
}

// ---------------------------------------------------------------------------
// Kernel 3: layer 2 + softmax + Jacobian sandwich. One wave per batch item.
//   mu3[o]   = sum_h w_mu2[h,o]*m2[b,h] + b_mu2[o]
//   quad2[o] = sum_h Ws2[o,h]*m2[b,h]^2
//   mid[o,p] = sum_h d2[b,h]*w_mu2[h,o]*w_mu2[h,p]   (symmetric, 55 uniques)
//   tr[b,o]  = Ws2[o',2c']*d2[c',c'] + Ws2[o',2c'+1]*d2[c',256+c']
//              with flat=b*10+o, o'=flat>>8, c'=flat&255   (torch-view replica)
//   sigma3   = mid + diag(tr + quad2 + softplus(b_sigma2))
//   p = softmax(mu3);  Sigma_out = J sigma3 J^T,  J = diag(p) - p p^T
// ---------------------------------------------------------------------------
__global__ __launch_bounds__(32) void layer2_softmax_kernel(
    const float* __restrict__ m2,       // (256,512)
    const float* __restrict__ d2,       // (256,512)
    const float* __restrict__ w_mu2,    // (512,10)
    const float* __restrict__ ws2,      // (10,512) softplus'd
    const float* __restrict__ b_mu2,    // (10)
    const float* __restrict__ b_sigma2, // (10)
    float* __restrict__ out_p,          // (256,10)
    float* __restrict__ out_S)          // (256,10,10)
{
    const int b    = blockIdx.x;
    const int lane = threadIdx.x;       // 0..31 (wave32)

    float mu3a[DOUT], q2a[DOUT], mid[55];
    #pragma unroll
    for (int o = 0; o < DOUT; ++o) { mu3a[o] = 0.0f; q2a[o] = 0.0f; }
    #pragma unroll
    for (int t = 0; t < 55; ++t) mid[t] = 0.0f;

    for (int h = lane; h < DH; h += 32) {
        const float m  = m2[(size_t)b * DH + h];
        const float d  = d2[(size_t)b * DH + h];
        const float mm = m * m;
        float w[DOUT];
        #pragma unroll
        for (int o = 0; o < DOUT; ++o) w[o] = w_mu2[(size_t)h * DOUT + o];
        #pragma unroll
        for (int o = 0; o < DOUT; ++o) {
            mu3a[o] += w[o] * m;
            q2a[o]  += ws2[o * DH + h] * mm;
        }
        int idx = 0;
        #pragma unroll
        for (int o = 0; o < DOUT; ++o) {
            const float dw = d * w[o];
            #pragma unroll
            for (int p = o; p < DOUT; ++p) { mid[idx] += dw * w[p]; ++idx; }
        }
    }

    // wave32 xor reductions (all lanes end with full sums)
    #pragma unroll
    for (int off = 16; off > 0; off >>= 1) {
        #pragma unroll
        for (int o = 0; o < DOUT; ++o) {
            mu3a[o] += __shfl_xor(mu3a[o], off, 32);
            q2a[o]  += __shfl_xor(q2a[o],  off, 32);
        }
        #pragma unroll
        for (int t = 0; t < 55; ++t) mid[t] += __shfl_xor(mid[t], off, 32);
    }

    __shared__ float sS[DOUT][DOUT];
    __shared__ float sT[DOUT][DOUT];
    __shared__ float sP[DOUT];
    __shared__ float sDp[DOUT];
    __shared__ float sTp[DOUT];

    if (lane == 0) {
        float mu3[DOUT];
        #pragma unroll
        for (int o = 0; o < DOUT; ++o) mu3[o] = mu3a[o] + b_mu2[o];

        int idx = 0;
        #pragma unroll
        for (int o = 0; o < DOUT; ++o) {
            #pragma unroll
            for (int p = o; p < DOUT; ++p) {
                const float v = mid[idx]; ++idx;
                sS[o][p] = v;
                sS[p][o] = v;
            }
        }
        #pragma unroll
        for (int o = 0; o < DOUT; ++o) {
            const int flat = b * DOUT + o;
            const int o2 = flat >> 8;
            const int c2 = flat & 255;
            const float tr =
                ws2[o2 * DH + 2 * c2]     * d2[(size_t)c2 * DH + c2] +
                ws2[o2 * DH + 2 * c2 + 1] * d2[(size_t)c2 * DH + 256 + c2];
            sS[o][o] += tr + q2a[o] + softplus_f(b_sigma2[o]);
        }

        // softmax
        float mx = mu3[0];
        #pragma unroll
        for (int o = 1; o < DOUT; ++o) mx = fmaxf(mx, mu3[o]);
        float sum = 0.0f;
        #pragma unroll
        for (int o = 0; o < DOUT; ++o) { const float e = expf(mu3[o] - mx); sP[o] = e; sum += e; }
        const float inv = 1.0f / sum;
        #pragma unroll
        for (int o = 0; o < DOUT; ++o) sP[o] *= inv;
    }
    __syncthreads();

    if (lane < DOUT) out_p[(size_t)b * DOUT + lane] = sP[lane];

    // sDp[k] = sum_j p_j * S[j][k]
    if (lane < DOUT) {
        float dp = 0.0f;
        #pragma unroll
        for (int j = 0; j < DOUT; ++j) dp += sP[j] * sS[j][lane];
        sDp[lane] = dp;
    }
    __syncthreads();

    // T[i][k] = p_i*(S[i][k] - sDp[k]);  sTp[i] = sum_k T[i][k]*p_k
    if (lane < DOUT) {
        const int i = lane;
        float tp = 0.0f;
        #pragma unroll
        for (int k = 0; k < DOUT; ++k) {
            const float t = sP[i] * (sS[i][k] - sDp[k]);
            sT[i][k] = t;
            tp += t * sP[k];
        }
        sTp[i] = tp;
    }
    __syncthreads();

    // Sigma_out[i][l] = p_l*(T[i][l] - sTp[i])
    if (lane < DOUT) {
        const int i = lane;
        #pragma unroll
        for (int l = 0; l < DOUT; ++l) {
            out_S[(size_t)b * DOUT * DOUT + i * DOUT + l] =
                sP[l] * (sT[i][l] - sTp[i]);
        }
    }
}

// ---------------------------------------------------------------------------
extern "C" void kernel_launch(void* const* d_in, const int* in_sizes, int n_in,
                              void* d_out, int out_size, void* d_ws, size_t ws_size,
                              hipStream_t stream) {
    const float* x        = (const float*)d_in[0];
    const float* w_mu1    = (const float*)d_in[1];
    const float* w_sigma1 = (const float*)d_in[2];
    const float* b_mu1    = (const float*)d_in[3];
    const float* b_sigma1 = (const float*)d_in[4];
    const float* w_mu2    = (const float*)d_in[5];
    const float* w_sigma2 = (const float*)d_in[6];
    const float* b_mu2    = (const float*)d_in[7];
    const float* b_sigma2 = (const float*)d_in[8];

    float* out   = (float*)d_out;
    float* out_p = out;                                   // 256*10
    float* out_S = out + BATCH * DOUT;                    // 256*100
    float* out_kl = out + BATCH * DOUT + BATCH * DOUT * DOUT;  // scalar

    float* ws  = (float*)d_ws;
    float* ws1 = ws;                         // 512*784 = 401408
    float* ws2 = ws1 + DH * DIN;             // 10*512  = 5120
    float* m2  = ws2 + DOUT * DH;            // 256*512 = 131072
    float* d2  = m2 + BATCH * DH;            // 256*512 = 131072
                                             // total ~2.67 MB

    // zero the kl accumulator (graph-capture safe)
    (void)hipMemsetAsync(out_kl, 0, sizeof(float), stream);

    // K1: softplus materialization + KL reduction
    prep_kl_kernel<<<(DH * DIN + 255) / 256, 256, 0, stream>>>(
        w_mu1, w_sigma1, w_mu2, w_sigma2, ws1, ws2, out_kl);

    // K2: layer-1 dual WMMA GEMM (async LDS-staged A) + ReLU epilogue
    dim3 blk1(32, 4);
    dim3 grd1(DH / 64, BATCH / 16);   // (8, 16): 512 waves, one 16x16 tile each
    layer1_wmma_kernel<<<grd1, blk1, 0, stream>>>(
        x, w_mu1, ws1, b_mu1, b_sigma1, m2, d2);

    // K3: layer 2 + softmax + Jacobian sandwich (one wave per batch item)
    layer2_softmax_kernel<<<BATCH, 32, 0, stream>>>(
        m2, d2, w_mu2, ws2, b_mu2, b_sigma2, out_p, out_S);
}